// Flow_49546742727298
// MI455X (gfx1250) — compile-verified
//
#include <hip/hip_runtime.h>
#include <math.h>

typedef float v2f __attribute__((ext_vector_type(2)));
typedef float v8f __attribute__((ext_vector_type(8)));

#define DD 16      // state dim
#define HH 64      // hidden width
#define NSTEPS 2
#define WAVES 8    // waves per block (256 threads)
#define HS 68      // LDS row stride in floats (bank-conflict free on 64 banks)

// Branch-free tanh built on the hardware transcendentals:
//   tanh(x) = 1 - 2/(exp(2x)+1) = 1 - 2*rcp(exp2(2*log2(e)*x) + 1)
// v_exp_f32 / v_rcp_f32 are TRANS ops that co-execute with VALU; no EXEC
// manipulation (WMMA-friendly), saturates to +-1, propagates NaN.
__device__ __forceinline__ float tanh_fast(float x) {
    const float e = __builtin_amdgcn_exp2f(x * 2.88539008177792681472f); // 2*log2(e)
    return 1.0f - 2.0f * __builtin_amdgcn_rcpf(e + 1.0f);
}

// One wave handles one 16-sample tile.
// GEMM1: pre[16x64] = X[16x16] @ W1[0:16,64]   (4 N-tiles x 4 K-chunks of v_wmma_f32_16x16x4_f32)
// epilogue: pre += t*W1[16] + b1 ; h = tanh(pre) ; div = sum_j (1-h^2)*c_j
// GEMM2: v[16x16]  = h[16x64] @ W2[64x16]      (16 K-chunks), + b2
// x <- x + dt*v ; logq <- logq - dt*div
__global__ __launch_bounds__(32 * WAVES)
void flow_kernel(const float* __restrict__ x0,
                 const float* __restrict__ W1, const float* __restrict__ b1,
                 const float* __restrict__ W2, const float* __restrict__ b2,
                 float* __restrict__ out, int batch)
{
    __shared__ float lds[WAVES * 16 * HS];
    const int lane = threadIdx.x & 31;
    const int wave = threadIdx.x >> 5;
    const int l15  = lane & 15;
    const int half = lane >> 4;          // 0 or 1 (upper half-wave)
    const int koff = half * 2;           // K offset inside a 4-wide K chunk
    float* sm = &lds[wave * 16 * HS];

    const int tile0 = (blockIdx.x * WAVES + wave) * 16;   // first batch row of this tile

    // ---------------- preload weights into WMMA B-operand layout ----------------
    // B (4x16 chunk): VGPR0 = rows {k0, k0+2} (half 0/1), VGPR1 = rows {k0+1, k0+3}
    v2f w1b[4][4];                       // [n_tile][k_chunk] of W1[0:16, 64]
    #pragma unroll
    for (int nt = 0; nt < 4; ++nt) {
        #pragma unroll
        for (int kc = 0; kc < 4; ++kc) {
            const int k = kc * 4 + koff;
            const int j = nt * 16 + l15;
            w1b[nt][kc].x = W1[(k    ) * HH + j];
            w1b[nt][kc].y = W1[(k + 1) * HH + j];
        }
    }
    v2f w2b[16];                         // W2[64x16], 16 K-chunks
    #pragma unroll
    for (int kc = 0; kc < 16; ++kc) {
        const int k = kc * 4 + koff;
        w2b[kc].x = W2[(k    ) * DD + l15];
        w2b[kc].y = W2[(k + 1) * DD + l15];
    }
    // per-lane epilogue constants for hidden index j = nt*16 + l15
    float w1t[4], bb1[4], cv[4];
    #pragma unroll
    for (int nt = 0; nt < 4; ++nt) {
        const int j = nt * 16 + l15;
        w1t[nt] = W1[16 * HH + j];       // time row of W1
        bb1[nt] = b1[j];
        float c = 0.f;                   // c_j = sum_i W1[i,j]*W2[j,i]  (trace weights)
        #pragma unroll
        for (int i = 0; i < 16; ++i) c += W1[i * HH + j] * W2[j * DD + i];
        cv[nt] = c;
    }
    const float b2v = b2[l15];

    // ---------------- load x tile in both layouts ----------------
    // A layout: lane holds row (l15), K = kc*4 + koff .. +1
    v2f xa[4];
    #pragma unroll
    for (int kc = 0; kc < 4; ++kc) {
        const float* p = &x0[(tile0 + l15) * DD + kc * 4 + koff];
        xa[kc].x = p[0];
        xa[kc].y = p[1];
    }
    // C layout: VGPR r holds row r + 8*half, col l15
    float xc[8];
    #pragma unroll
    for (int r = 0; r < 8; ++r)
        xc[r] = x0[(tile0 + r + half * 8) * DD + l15];

    float dlq[8];
    #pragma unroll
    for (int r = 0; r < 8; ++r) dlq[r] = 0.f;

    const float dt = 1.0f / (float)NSTEPS;

    #pragma unroll
    for (int step = 0; step < NSTEPS; ++step) {
        const float t = (float)step * dt;

        // ---- GEMM1: pre = X @ W1[0:16,:]  (fp32 WMMA) ----
        v8f acc[4];
        #pragma unroll
        for (int nt = 0; nt < 4; ++nt) {
            v8f c = {};
            #pragma unroll
            for (int kc = 0; kc < 4; ++kc)
                c = __builtin_amdgcn_wmma_f32_16x16x4_f32(
                        false, xa[kc], false, w1b[nt][kc], (short)0, c, false, false);
            acc[nt] = c;
        }

        // ---- epilogue: h = tanh(pre + t*W1[16] + b1); divergence partials; h -> LDS ----
        float p[8];
        #pragma unroll
        for (int r = 0; r < 8; ++r) p[r] = 0.f;
        #pragma unroll
        for (int nt = 0; nt < 4; ++nt) {
            #pragma unroll
            for (int r = 0; r < 8; ++r) {
                const float hh = tanh_fast(acc[nt][r] + t * w1t[nt] + bb1[nt]);
                p[r] += (1.f - hh * hh) * cv[nt];
                sm[(r + half * 8) * HS + nt * 16 + l15] = hh;
            }
        }
        // reduce trace partials across the 16-lane half (xor 1,2,4,8 stays in-half)
        #pragma unroll
        for (int r = 0; r < 8; ++r) {
            float v = p[r];
            v += __shfl_xor(v, 1);
            v += __shfl_xor(v, 2);
            v += __shfl_xor(v, 4);
            v += __shfl_xor(v, 8);
            dlq[r] -= dt * v;            // sample index: r + 8*half
        }
        __syncthreads();

        // ---- GEMM2: v = h @ W2  (h read back from LDS in A layout) ----
        v8f vacc = {};
        #pragma unroll
        for (int kc = 0; kc < 16; ++kc) {
            const int k = kc * 4 + koff;
            v2f a;
            a.x = sm[l15 * HS + k];
            a.y = sm[l15 * HS + k + 1];
            vacc = __builtin_amdgcn_wmma_f32_16x16x4_f32(
                       false, a, false, w2b[kc], (short)0, vacc, false, false);
        }
        __syncthreads();

        // ---- Euler update in C layout ----
        #pragma unroll
        for (int r = 0; r < 8; ++r)
            xc[r] += dt * (vacc[r] + b2v);

        if (step + 1 < NSTEPS) {
            // bounce updated x back to A layout through LDS for the next step
            #pragma unroll
            for (int r = 0; r < 8; ++r)
                sm[(r + half * 8) * HS + l15] = xc[r];
            __syncthreads();
            #pragma unroll
            for (int kc = 0; kc < 4; ++kc) {
                xa[kc].x = sm[l15 * HS + kc * 4 + koff];
                xa[kc].y = sm[l15 * HS + kc * 4 + koff + 1];
            }
            __syncthreads();
        }
    }

    // ---------------- store x1 [batch,16] then dlogq [batch] ----------------
    float* __restrict__ x1 = out;
    float* __restrict__ lq = out + (size_t)batch * DD;
    #pragma unroll
    for (int r = 0; r < 8; ++r)
        x1[(tile0 + r + half * 8) * DD + l15] = xc[r];
    #pragma unroll
    for (int r = 0; r < 8; ++r)
        if (l15 == r)                    // lanes r and 16+r hold dlq for rows r and r+8
            lq[tile0 + r + half * 8] = dlq[r];
}

extern "C" void kernel_launch(void* const* d_in, const int* in_sizes, int n_in,
                              void* d_out, int out_size, void* d_ws, size_t ws_size,
                              hipStream_t stream) {
    const float* x0 = (const float*)d_in[0];
    const float* W1 = (const float*)d_in[1];
    const float* b1 = (const float*)d_in[2];
    const float* W2 = (const float*)d_in[3];
    const float* b2 = (const float*)d_in[4];
    float* out = (float*)d_out;

    const int batch  = in_sizes[0] / DD;          // 131072
    const int blocks = batch / (16 * WAVES);      // 1024 blocks of 256 threads

    flow_kernel<<<blocks, 32 * WAVES, 0, stream>>>(x0, W1, b1, W2, b2, out, batch);
}